// DeepGAT_64785286692909
// MI455X (gfx1250) — compile-verified
//
#include <hip/hip_runtime.h>
#include <hip/hip_bf16.h>

typedef __attribute__((ext_vector_type(2))) float v2f;
typedef __attribute__((ext_vector_type(8))) float v8f;

#define DH 32
#define NEG_BIG (-3.0e38f)

// ---------------- wave helpers (wave32) ----------------
__device__ __forceinline__ float waveSum(float v) {
#pragma unroll
  for (int m = 16; m >= 1; m >>= 1) v += __shfl_xor(v, m, 32);
  return v;
}

__device__ __forceinline__ void atomicMaxF(float* addr, float val) {
  int* ai = (int*)addr;
  int old = __float_as_int(*addr);
  while (__int_as_float(old) < val) {
    int assumed = old;
    old = atomicCAS(ai, assumed, __float_as_int(val));
    if (old == assumed) break;
  }
}

// ---------------- generic fill ----------------
__global__ void k_fill(float* __restrict__ p, float v, long long n) {
  long long t = (long long)blockIdx.x * blockDim.x + threadIdx.x;
  if (t < n) p[t] = v;
}

// ---------------- degree / self-loop attr ----------------
__global__ void k_deg(const int* __restrict__ dst, const float* __restrict__ ea,
                      float* __restrict__ degsum, float* __restrict__ cnt, int E) {
  long long t = (long long)blockIdx.x * blockDim.x + threadIdx.x;
  if (t >= E) return;
  int d = dst[t];
  atomicAdd(&degsum[d], ea[t]);
  atomicAdd(&cnt[d], 1.0f);
}

__global__ void k_prep(const float* __restrict__ degsum, const float* __restrict__ cnt,
                       float* __restrict__ dinv, float* __restrict__ lat, int N) {
  long long t = (long long)blockIdx.x * blockDim.x + threadIdx.x;
  if (t >= N) return;
  float s = degsum[t];
  dinv[t] = rsqrtf(s + 1.0f);              // +1 = self-loop weight; always > 0
  lat[t]  = s / fmaxf(cnt[t], 1.0f);       // mean incoming edge_attr
}

// ---------------- input projection (K=2, scalar) ----------------
__global__ void k_in_proj(const float* __restrict__ x,
                          const float* __restrict__ W1, const float* __restrict__ P1,
                          float* __restrict__ hW, float* __restrict__ proj, int N) {
  long long t = (long long)blockIdx.x * blockDim.x + threadIdx.x;
  int n = (int)(t >> 5), c = (int)(t & 31);
  if (n >= N) return;
  float x0 = x[(size_t)n * 2], x1 = x[(size_t)n * 2 + 1];
  hW[(size_t)n * DH + c]   = x0 * W1[c] + x1 * W1[DH + c];
  proj[(size_t)n * DH + c] = x0 * P1[c] + x1 * P1[DH + c];
}

// ---------------- GCN edge scatter ----------------
__global__ void k_gcn_scatter(const float* __restrict__ hW,
                              const int* __restrict__ src, const int* __restrict__ dst,
                              const float* __restrict__ ea, const float* __restrict__ dinv,
                              float* __restrict__ agg, int E) {
  long long t = (long long)blockIdx.x * blockDim.x + threadIdx.x;
  int e = (int)(t >> 5), c = (int)(t & 31);
  if (e >= E) return;
  if (c == 0 && e + 4096 < E) {            // prefetch the sequential index/attr streams
    __builtin_prefetch(&src[e + 4096], 0, 1);
    __builtin_prefetch(&dst[e + 4096], 0, 1);
    __builtin_prefetch(&ea[e + 4096], 0, 1);
  }
  int s = src[e], d = dst[e];
  float nrm = dinv[s] * ea[e] * dinv[d];
  atomicAdd(&agg[(size_t)d * DH + c], hW[(size_t)s * DH + c] * nrm);
}

// ---------------- fused (self-loop) + bias + residual + LayerNorm + ELU ----------------
__global__ void k_ln_elu(const float* __restrict__ agg,
                         const float* __restrict__ selfH, const float* __restrict__ dinv,
                         const float* __restrict__ b, const float* __restrict__ proj,
                         const float* __restrict__ pb, const float* __restrict__ g,
                         const float* __restrict__ be, float* __restrict__ out, int N) {
  long long t = (long long)blockIdx.x * blockDim.x + threadIdx.x;
  int n = (int)(t >> 5), c = (int)(t & 31);
  if (n >= N) return;
  float v = agg[(size_t)n * DH + c] + b[c] + proj[(size_t)n * DH + c] + pb[c];
  if (selfH) { float di = dinv[n]; v += selfH[(size_t)n * DH + c] * di * di; }
  float mean = waveSum(v) * (1.0f / 32.0f);
  float dv = v - mean;
  float var = waveSum(dv * dv) * (1.0f / 32.0f);
  float y = dv * rsqrtf(var + 1e-5f) * g[c] + be[c];
  out[(size_t)n * DH + c] = (y > 0.0f) ? y : (expf(y) - 1.0f);
}

// ---------------- WMMA dual GEMM (full fp32, V_WMMA_F32_16X16X4_F32) ----------------
// Ya = X@Wa(+ba), Yb = X@Wb(+bb).  X: N x 32 fp32, W: 32 x 32 fp32 row-major.
// One 16-node tile per wave; weights staged transposed in LDS once per block.
__global__ void k_wmma_dual(const float* __restrict__ X,
                            const float* __restrict__ Wa, const float* __restrict__ ba,
                            float* __restrict__ Ya,
                            const float* __restrict__ Wb, const float* __restrict__ bb,
                            float* __restrict__ Yb,
                            int N, int doRelu) {
  __shared__ float WT[2][DH][DH];          // [matrix][col][k]  (transposed), 8 KB
  const int tid = threadIdx.x;
  for (int i = tid; i < DH * DH; i += blockDim.x) {
    int k = i >> 5, c = i & 31;            // i = k*32 + c (row-major source)
    WT[0][c][k] = Wa[i];
    WT[1][c][k] = Wb ? Wb[i] : 0.0f;
  }
  __syncthreads();

  const int lane = tid & 31;
  const int tile = blockIdx.x * (blockDim.x >> 5) + (tid >> 5);
  if (tile * 16 >= N) return;              // wave-uniform, after the barrier

  const int row  = lane & 15;
  const float* xr = X + (size_t)(tile * 16 + row) * DH;
  const int koff = (lane < 16) ? 0 : 2;    // A/B 32-bit layout: K pair per half-wave
  const int col  = lane & 15;

  v8f a0 = {}, a1 = {}, b0 = {}, b1 = {};  // [Wa|Wb] x [cols 0-15 | cols 16-31]
#pragma unroll
  for (int kk = 0; kk < DH; kk += 4) {
    v2f av;  av[0]  = xr[kk + koff];             av[1]  = xr[kk + koff + 1];
    v2f bv0; bv0[0] = WT[0][col][kk + koff];      bv0[1] = WT[0][col][kk + koff + 1];
    v2f bv1; bv1[0] = WT[0][col + 16][kk + koff]; bv1[1] = WT[0][col + 16][kk + koff + 1];
    a0 = __builtin_amdgcn_wmma_f32_16x16x4_f32(false, av, false, bv0, (short)0, a0, false, false);
    a1 = __builtin_amdgcn_wmma_f32_16x16x4_f32(false, av, false, bv1, (short)0, a1, false, false);
    if (Wb) {
      v2f bv2; bv2[0] = WT[1][col][kk + koff];      bv2[1] = WT[1][col][kk + koff + 1];
      v2f bv3; bv3[0] = WT[1][col + 16][kk + koff]; bv3[1] = WT[1][col + 16][kk + koff + 1];
      b0 = __builtin_amdgcn_wmma_f32_16x16x4_f32(false, av, false, bv2, (short)0, b0, false, false);
      b1 = __builtin_amdgcn_wmma_f32_16x16x4_f32(false, av, false, bv3, (short)0, b1, false, false);
    }
  }

  const int mBase = (lane < 16) ? 0 : 8;   // 32-bit C/D layout
  const float ba0 = ba ? ba[col] : 0.0f;
  const float ba1 = ba ? ba[col + 16] : 0.0f;
#pragma unroll
  for (int r = 0; r < 8; ++r) {
    int nd = tile * 16 + mBase + r;
    float v0 = a0[r] + ba0, v1 = a1[r] + ba1;
    if (doRelu) { v0 = fmaxf(v0, 0.0f); v1 = fmaxf(v1, 0.0f); }
    Ya[(size_t)nd * DH + col]      = v0;
    Ya[(size_t)nd * DH + col + 16] = v1;
  }
  if (Wb) {
    const float bb0 = bb ? bb[col] : 0.0f;
    const float bb1 = bb ? bb[col + 16] : 0.0f;
#pragma unroll
    for (int r = 0; r < 8; ++r) {
      int nd = tile * 16 + mBase + r;
      float v0 = b0[r] + bb0, v1 = b1[r] + bb1;
      if (doRelu) { v0 = fmaxf(v0, 0.0f); v1 = fmaxf(v1, 0.0f); }
      Yb[(size_t)nd * DH + col]      = v0;
      Yb[(size_t)nd * DH + col + 16] = v1;
    }
  }
}

// ---------------- GATv2 passes ----------------
__global__ void k_gat_logits(const float* __restrict__ xl, const float* __restrict__ xr,
                             const float* __restrict__ We, const float* __restrict__ att,
                             const int* __restrict__ src, const int* __restrict__ dst,
                             const float* __restrict__ ea, const float* __restrict__ lat,
                             float* __restrict__ logits, float* __restrict__ mx, int E, int N) {
  long long t = (long long)blockIdx.x * blockDim.x + threadIdx.x;
  int e = (int)(t >> 5), c = (int)(t & 31);
  if (e >= E + N) return;
  if (c == 0 && e + 4096 < E) {
    __builtin_prefetch(&src[e + 4096], 0, 1);
    __builtin_prefetch(&dst[e + 4096], 0, 1);
  }
  int s, d; float aat;
  if (e < E) { s = src[e]; d = dst[e]; aat = ea[e]; }
  else       { s = d = e - E;          aat = lat[e - E]; }
  float m = xl[(size_t)s * DH + c] + xr[(size_t)d * DH + c] + aat * We[c];
  m = (m > 0.0f) ? m : 0.2f * m;               // leaky_relu(0.2)
  float l = waveSum(m * att[c]);
  if (c == 0) { logits[e] = l; atomicMaxF(&mx[d], l); }
}

__global__ void k_gat_exp(float* __restrict__ logits, const float* __restrict__ mx,
                          float* __restrict__ denom, const int* __restrict__ dst, int E, int N) {
  long long t = (long long)blockIdx.x * blockDim.x + threadIdx.x;
  if (t >= E + N) return;
  int e = (int)t;
  int d = (e < E) ? dst[e] : e - E;
  float ex = expf(logits[e] - mx[d]);
  logits[e] = ex;
  atomicAdd(&denom[d], ex);
}

__global__ void k_gat_alpha(const float* __restrict__ logits, const float* __restrict__ denom,
                            const float* __restrict__ xl,
                            const int* __restrict__ src, const int* __restrict__ dst,
                            float* __restrict__ alphaOut, float* __restrict__ agg, int E, int N) {
  long long t = (long long)blockIdx.x * blockDim.x + threadIdx.x;
  int e = (int)(t >> 5), c = (int)(t & 31);
  if (e >= E + N) return;
  int s, d;
  if (e < E) { s = src[e]; d = dst[e]; } else { s = d = e - E; }
  float alpha = logits[e] / denom[d];
  if (c == 0) alphaOut[e] = alpha;
  atomicAdd(&agg[(size_t)d * DH + c], alpha * xl[(size_t)s * DH + c]);
}

// ---------------- classifier heads + grouped log-softmax ----------------
__global__ void k_heads(const float* __restrict__ ht1, const float* __restrict__ hc1,
                        const float* __restrict__ Ct2, const float* __restrict__ ct2b,
                        const float* __restrict__ Cc2, const float* __restrict__ cc2b,
                        float* __restrict__ out, int N) {
  long long t = (long long)blockIdx.x * blockDim.x + threadIdx.x;
  int n = (int)(t >> 5), j = (int)(t & 31);
  if (n >= N) return;
  float v = NEG_BIG;
  if (j < 19) {
    float acc = cc2b[j];
    const float* h = hc1 + (size_t)n * DH;
#pragma unroll
    for (int k = 0; k < DH; ++k) acc += h[k] * Cc2[k * 19 + j];
    v = acc;
  } else if (j < 28) {
    int jj = j - 19;
    float acc = ct2b[jj];
    const float* h = ht1 + (size_t)n * DH;
#pragma unroll
    for (int k = 0; k < DH; ++k) acc += h[k] * Ct2[k * 9 + jj];
    v = acc;
  }
  float mc = NEG_BIG, mt = NEG_BIG;
  for (int i = 0; i < 19; ++i)  mc = fmaxf(mc, __shfl(v, i, 32));
  for (int i = 19; i < 28; ++i) mt = fmaxf(mt, __shfl(v, i, 32));
  float sc = 0.0f, st = 0.0f;
  for (int i = 0; i < 19; ++i)  sc += expf(__shfl(v, i, 32) - mc);
  for (int i = 19; i < 28; ++i) st += expf(__shfl(v, i, 32) - mt);
  if (j < 19)       out[(size_t)n * 28 + j] = v - mc - logf(sc);
  else if (j < 28)  out[(size_t)n * 28 + j] = v - mt - logf(st);
}

// ---------------- host orchestration ----------------
extern "C" void kernel_launch(void* const* d_in, const int* in_sizes, int n_in,
                              void* d_out, int out_size, void* d_ws, size_t ws_size,
                              hipStream_t stream) {
  const float* x   = (const float*)d_in[0];
  const int*   ei  = (const int*)d_in[1];
  const float* ea  = (const float*)d_in[2];
  const float* W1  = (const float*)d_in[3];
  const float* b1  = (const float*)d_in[4];
  const float* P1  = (const float*)d_in[5];
  const float* pb1 = (const float*)d_in[6];
  const float* g1  = (const float*)d_in[7];
  const float* be1 = (const float*)d_in[8];
  const float* W2  = (const float*)d_in[9];
  const float* b2  = (const float*)d_in[10];
  const float* P2  = (const float*)d_in[11];
  const float* pb2 = (const float*)d_in[12];
  const float* g2  = (const float*)d_in[13];
  const float* be2 = (const float*)d_in[14];
  const float* Wl  = (const float*)d_in[15];
  const float* bl  = (const float*)d_in[16];
  const float* Wr  = (const float*)d_in[17];
  const float* br  = (const float*)d_in[18];
  const float* We  = (const float*)d_in[19];
  const float* att = (const float*)d_in[20];
  const float* gb  = (const float*)d_in[21];
  const float* P3  = (const float*)d_in[22];
  const float* pb3 = (const float*)d_in[23];
  const float* g3  = (const float*)d_in[24];
  const float* be3 = (const float*)d_in[25];
  const float* Ct1 = (const float*)d_in[26];
  const float* ct1b= (const float*)d_in[27];
  const float* Cc1 = (const float*)d_in[28];
  const float* cc1b= (const float*)d_in[29];
  const float* Ct2 = (const float*)d_in[30];
  const float* ct2b= (const float*)d_in[31];
  const float* Cc2 = (const float*)d_in[32];
  const float* cc2b= (const float*)d_in[33];

  const int N = in_sizes[0] / 2;
  const int E = in_sizes[1] / 2;
  const int* src = ei;
  const int* dst = ei + E;

  float* w      = (float*)d_ws;
  float* degsum = w;                               // N   (later: mx)
  float* cnt    = w + (size_t)N;                   // N   (later: denom)
  float* dinv   = w + 2 * (size_t)N;               // N
  float* lat    = w + 3 * (size_t)N;               // N
  float* B0     = w + 4 * (size_t)N;               // N*32  hW
  float* B1     = B0 + (size_t)N * DH;             // N*32  proj
  float* B2     = B1 + (size_t)N * DH;             // N*32  agg
  float* B3     = B2 + (size_t)N * DH;             // N*32  x1 -> x3
  float* B4     = B3 + (size_t)N * DH;             // N*32  x2
  float* B5     = B4 + (size_t)N * DH;             // N*32  xl -> hc1
  float* B6     = B5 + (size_t)N * DH;             // N*32  xr -> ht1
  float* logits = B6 + (size_t)N * DH;             // E+N
  float* mx     = degsum;
  float* denom  = cnt;
  float* outv   = (float*)d_out;
  float* alphaOut = outv + (size_t)N * 28;

  const int BS = 256;
  auto nb = [](long long t) { return (unsigned)((t + 255) / 256); };
  const long long nwaveN = (long long)((N + 15) / 16) * 32;   // WMMA: 1 wave / 16 rows

  // degrees + self-loop attr
  k_fill<<<nb(4LL * N), BS, 0, stream>>>(w, 0.0f, 4LL * N);
  k_deg<<<nb(E), BS, 0, stream>>>(dst, ea, degsum, cnt, E);
  k_prep<<<nb(N), BS, 0, stream>>>(degsum, cnt, dinv, lat, N);

  // ---- GCN layer 1 ----
  k_in_proj<<<nb((long long)N * DH), BS, 0, stream>>>(x, W1, P1, B0, B1, N);
  k_fill<<<nb((long long)N * DH), BS, 0, stream>>>(B2, 0.0f, (long long)N * DH);
  k_gcn_scatter<<<nb((long long)E * DH), BS, 0, stream>>>(B0, src, dst, ea, dinv, B2, E);
  k_ln_elu<<<nb((long long)N * DH), BS, 0, stream>>>(B2, B0, dinv, b1, B1, pb1, g1, be1, B3, N);

  // ---- GCN layer 2 (WMMA) ----
  k_wmma_dual<<<nb(nwaveN), BS, 0, stream>>>(B3, W2, nullptr, B0, P2, nullptr, B1, N, 0);
  k_fill<<<nb((long long)N * DH), BS, 0, stream>>>(B2, 0.0f, (long long)N * DH);
  k_gcn_scatter<<<nb((long long)E * DH), BS, 0, stream>>>(B0, src, dst, ea, dinv, B2, E);
  k_ln_elu<<<nb((long long)N * DH), BS, 0, stream>>>(B2, B0, dinv, b2, B1, pb2, g2, be2, B4, N);

  // ---- GATv2 (WMMA for Wl/Wr/P3) ----
  k_wmma_dual<<<nb(nwaveN), BS, 0, stream>>>(B4, Wl, bl, B5, Wr, br, B6, N, 0);
  k_wmma_dual<<<nb(nwaveN), BS, 0, stream>>>(B4, P3, nullptr, B1, nullptr, nullptr, nullptr, N, 0);
  k_fill<<<nb(N), BS, 0, stream>>>(mx, NEG_BIG, N);
  k_fill<<<nb(N), BS, 0, stream>>>(denom, 0.0f, N);
  k_fill<<<nb((long long)N * DH), BS, 0, stream>>>(B2, 0.0f, (long long)N * DH);
  k_gat_logits<<<nb((long long)(E + N) * 32), BS, 0, stream>>>(B5, B6, We, att, src, dst, ea, lat,
                                                               logits, mx, E, N);
  k_gat_exp<<<nb((long long)(E + N)), BS, 0, stream>>>(logits, mx, denom, dst, E, N);
  k_gat_alpha<<<nb((long long)(E + N) * 32), BS, 0, stream>>>(logits, denom, B5, src, dst,
                                                              alphaOut, B2, E, N);
  k_ln_elu<<<nb((long long)N * DH), BS, 0, stream>>>(B2, nullptr, nullptr, gb, B1, pb3, g3, be3, B3, N);

  // ---- classifier heads (WMMA + fused ReLU) ----
  k_wmma_dual<<<nb(nwaveN), BS, 0, stream>>>(B3, Ct1, ct1b, B6, Cc1, cc1b, B5, N, 1);
  k_heads<<<nb((long long)N * 32), BS, 0, stream>>>(B6, B5, Ct2, ct2b, Cc2, cc2b, outv, N);
}